// _Point2MaskModuleBase_87686052315593
// MI455X (gfx1250) — compile-verified
//
#include <hip/hip_runtime.h>
#include <hip/hip_bf16.h>

// CDNA5 / gfx1250, wave32.
typedef __attribute__((ext_vector_type(2))) float v2f;
typedef __attribute__((ext_vector_type(8))) float v8f;

#define KSEL 16
#define RAD2 4.0f
#define BIGF 3.0e38f
#define SENT 1.0e30f
#define SLAB 2048            // points staged per LDS slab (32 KB of float4)

// ---------------- Kernel 0: per-batch bbox (min/max over N points, both dims) --------
__global__ __launch_bounds__(256) void bbox_kernel(const float* __restrict__ coords,
                                                   float* __restrict__ bbox, int N) {
    int b = blockIdx.x;
    const float* c = coords + (size_t)b * N * 2;
    float mnx = BIGF, mxx = -BIGF, mny = BIGF, mxy = -BIGF;
    for (int n = threadIdx.x; n < N; n += blockDim.x) {
        float x = c[2 * n], y = c[2 * n + 1];
        mnx = fminf(mnx, x); mxx = fmaxf(mxx, x);
        mny = fminf(mny, y); mxy = fmaxf(mxy, y);
    }
    __shared__ float r0[256], r1[256], r2[256], r3[256];
    int t = threadIdx.x;
    r0[t] = mnx; r1[t] = mxx; r2[t] = mny; r3[t] = mxy;
    __syncthreads();
    for (int s = 128; s > 0; s >>= 1) {
        if (t < s) {
            r0[t] = fminf(r0[t], r0[t + s]);
            r1[t] = fmaxf(r1[t], r1[t + s]);
            r2[t] = fminf(r2[t], r2[t + s]);
            r3[t] = fmaxf(r3[t], r3[t + s]);
        }
        __syncthreads();
    }
    if (t == 0) {
        float cx = (r1[0] + r0[0]) * 0.5f, cy = (r3[0] + r2[0]) * 0.5f;
        float sx = fmaxf(r1[0] - r0[0], 1e-5f) * 0.5f;
        float sy = fmaxf(r3[0] - r2[0], 1e-5f) * 0.5f;
        bbox[b * 4 + 0] = cx; bbox[b * 4 + 1] = cy;
        bbox[b * 4 + 2] = sx; bbox[b * 4 + 3] = sy;
    }
}

// ---------------- Kernel 1: normalize coords -> Gram payload; top-2 features ---------
// payload row per point: [cx, cy, cx^2+cy^2, 1]   (A-matrix rows for WMMA)
// feat2 per point: (second_max, max) of the C feature channels
__global__ __launch_bounds__(256) void prep_kernel(const float* __restrict__ coords,
                                                   const float* __restrict__ features,
                                                   const float* __restrict__ bbox,
                                                   float4* __restrict__ payload,
                                                   float2* __restrict__ feat2,
                                                   int N, int C, int res, int totalP) {
    int p = blockIdx.x * blockDim.x + threadIdx.x;
    if (p >= totalP) return;
    int b = p / N;
    float cx = coords[2 * (size_t)p], cy = coords[2 * (size_t)p + 1];
    float ctrx = bbox[b * 4 + 0], ctry = bbox[b * 4 + 1];
    float scx  = bbox[b * 4 + 2], scy  = bbox[b * 4 + 3];
    float fr = (float)res;
    // ((c-center)/scale + 1) * 0.8*res/2 + 0.1*res
    float nx = ((cx - ctrx) / scx + 1.0f) * (0.4f * fr) + 0.1f * fr;
    float ny = ((cy - ctry) / scy + 1.0f) * (0.4f * fr) + 0.1f * fr;
    payload[p] = make_float4(nx, ny, nx * nx + ny * ny, 1.0f);
    const float* f = features + (size_t)p * C;
    float m1 = -BIGF, m2 = -BIGF;
    for (int j = 0; j < C; j++) {
        float v = f[j];
        if (v > m1) { m2 = m1; m1 = v; }
        else if (v > m2) { m2 = v; }
    }
    feat2[p] = make_float2(m2, m1); // channel0 = 2nd max, channel1 = max
}

// ---------------- Kernel 2: WMMA distance tiles + top-16 ball query + softmax --------
// Block = 8 waves; each wave owns a tile of 16 grid cells of ONE batch (grid.y = b).
// Payload is staged in 32KB LDS slabs shared by all 8 waves (8x L2 traffic cut).
// D = A x B via v_wmma_f32_16x16x4_f32:  A = points 16x4 [cx,cy,|c|^2,1],
// B = cells 4x16 [-2sx,-2sy,1,|s|^2]^T  =>  D[m][n] = exact squared distance.
// D layout (wave32): lane holds column n=lane&15; VGPR r holds row m=r+8*(lane>=16).
// Per-lane 16-entry replace-worst list; min-tree early-skip gates the candidate loop;
// ds_load double-buffer hides LDS latency behind the WMMA + selection work.
// The merge scratch overlays the slab buffer (needed only after the last slab).
__global__ __launch_bounds__(256) void ball_kernel(const float4* __restrict__ payload,
                                                   const float2* __restrict__ feat2,
                                                   const int* __restrict__ pnum_arr,
                                                   float* __restrict__ out,
                                                   int N, int res, int tilesPerBatch) {
    __shared__ __align__(16) unsigned char smem[SLAB * 16];  // 32 KB, dual purpose
    float4* spay = (float4*)smem;                    // slab staging (main loop)
    float*  sdp  = (float*)smem;                     // [8][32][16] merge dists
    int*    sip  = (int*)(smem + SLAB * 8);          // [8][32][16] merge indices

    const int w = threadIdx.x >> 5;
    const int lane = threadIdx.x & 31;
    const int b = blockIdx.y;
    const int tb = blockIdx.x * 8 + w;               // tile within batch
    const bool active = tb < tilesPerBatch;
    const int t0 = tb * 16;

    const float4* payB = payload + (size_t)b * N;

    // ---- per-wave constants -------------------------------------------------------
    const int n = lane & 15;
    const int cell = t0 + n;
    const float sx = (float)(cell / res);
    const float sy = (float)(cell % res);
    v2f bmat;
    if (lane < 16) { bmat[0] = -2.0f * sx; bmat[1] = -2.0f * sy; }       // K0,K1
    else           { bmat[0] = 1.0f;       bmat[1] = sx * sx + sy * sy; } // K2,K3
    const int pnum  = pnum_arr[b];
    const int koff  = (lane < 16) ? 0 : 2;   // A K-pair selection per lane half
    const int half8 = (lane >> 4) << 3;      // D row offset per lane half

    float bd[KSEL]; int bi[KSEL];
#pragma unroll
    for (int j = 0; j < KSEL; j++) { bd[j] = BIGF; bi[j] = 0; }
    float worst = BIGF; int wslot = 0;

    // ---- main loop over payload slabs --------------------------------------------
    for (int sb = 0; sb < N; sb += SLAB) {
        const int slabPts = min(SLAB, N - sb);
        __syncthreads();                              // previous slab fully consumed
        for (int i = threadIdx.x; i < SLAB; i += 256) {
            int gp = sb + i;
            spay[i] = (gp < N) ? payB[gp]
                               : make_float4(0.0f, 0.0f, SENT, 1.0f);  // far sentinel
        }
        __syncthreads();
        if (sb + SLAB < N)   // speculative prefetch of the next slab (one line each)
            __builtin_prefetch(payB + sb + SLAB + threadIdx.x * (SLAB / 256), 0, 3);

        if (active) {
            const int nch = (slabPts + 15) >> 4;      // 16-point chunks in this slab
            // register double-buffer of the A operand (ds_load_b64)
            const float* pf0 = (const float*)(spay + (lane & 15)) + koff;
            v2f a; a[0] = pf0[0]; a[1] = pf0[1];
            for (int c = 0; c < nch; c++) {
                v2f an = a;
                if (c + 1 < nch) {
                    const float* pf =
                        (const float*)(spay + ((c + 1) << 4) + (lane & 15)) + koff;
                    an[0] = pf[0]; an[1] = pf[1];
                }
                v8f cz = {};
                v8f d = __builtin_amdgcn_wmma_f32_16x16x4_f32(
                    false, a, false, bmat, (short)0, cz, false, false);
                // min-tree early skip: any candidate both in-radius and improving?
                float mn8 = fminf(fminf(fminf(d[0], d[1]), fminf(d[2], d[3])),
                                  fminf(fminf(d[4], d[5]), fminf(d[6], d[7])));
                if (mn8 <= RAD2 && mn8 < worst) {
#pragma unroll
                    for (int r = 0; r < 8; r++) {
                        float val = d[r];
                        int pidx = sb + (c << 4) + half8 + r;
                        if (val <= RAD2 && pidx < pnum && val < worst) {
                            // select-write keeps arrays in VGPRs (no scratch)
#pragma unroll
                            for (int j = 0; j < KSEL; j++)
                                if (j == wslot) { bd[j] = val; bi[j] = pidx; }
                            float mx = bd[0]; int ms = 0;
#pragma unroll
                            for (int j = 1; j < KSEL; j++)
                                if (bd[j] > mx) { mx = bd[j]; ms = j; }
                            worst = mx; wslot = ms;
                        }
                    }
                }
                a = an;
            }
        }
    }

    // ---- merge lanes L / L+16 through the (now free) LDS buffer -------------------
    __syncthreads();                                  // slab buffer dead, safe to reuse
    if (active) {
#pragma unroll
        for (int j = 0; j < KSEL; j++) {
            sdp[(w * 32 + lane) * KSEL + j] = bd[j];
            sip[(w * 32 + lane) * KSEL + j] = bi[j];
        }
    }
    // reads below only touch data written by this same wave (in-order DS pipeline)
    if (active && lane < 16) {
        float cd[32]; int ci[32];
#pragma unroll
        for (int j = 0; j < KSEL; j++) {
            cd[j]      = sdp[(w * 32 + lane) * KSEL + j];
            ci[j]      = sip[(w * 32 + lane) * KSEL + j];
            cd[j + 16] = sdp[(w * 32 + lane + 16) * KSEL + j];
            ci[j + 16] = sip[(w * 32 + lane + 16) * KSEL + j];
        }
        const float2* f2 = feat2 + (size_t)b * N;
        float s0a = 0.0f, s1a = 0.0f; int cnt = 0;
        for (int k = 0; k < KSEL; k++) {
            float mn = cd[0]; int arg = 0;
#pragma unroll
            for (int j = 1; j < 32; j++)
                if (cd[j] < mn) { mn = cd[j]; arg = j; }
            if (!(mn <= RAD2)) break;                 // no more valid candidates
            int pick = 0;
#pragma unroll
            for (int j = 0; j < 32; j++) if (j == arg) pick = ci[j];
            float2 f = f2[pick];
            s0a += f.x; s1a += f.y; cnt++;
#pragma unroll
            for (int j = 0; j < 32; j++) if (j == arg) cd[j] = BIGF;
        }
        float occ = (cnt > 0) ? (float)cnt : 1.0f;
        float nf0 = s0a / occ, nf1 = s1a / occ;
        float m  = fmaxf(nf0, nf1);
        float e0 = __expf(nf0 - m), e1 = __expf(nf1 - m);
        float inv = 1.0f / (e0 + e1);
        float p0 = e0 * inv, p1 = e1 * inv;
        if (p0 == p1) { p0 = 1.0f; p1 = 0.0f; }      // empty / degenerate cell
        size_t o = ((size_t)b * (size_t)(res * res) + (size_t)(t0 + lane)) * 2;
        out[o]     = p0;
        out[o + 1] = p1;
    }
}

// ---------------- Host launcher ------------------------------------------------------
extern "C" void kernel_launch(void* const* d_in, const int* in_sizes, int n_in,
                              void* d_out, int out_size, void* d_ws, size_t ws_size,
                              hipStream_t stream) {
    const float* coords     = (const float*)d_in[0];
    const float* features   = (const float*)d_in[1];
    const int*   points_num = (const int*)d_in[2];
    // d_in[3] is `res` on device; host derives it from out_size (no device reads).

    const int B  = in_sizes[2];
    const int N  = in_sizes[0] / (2 * B);
    const int C  = in_sizes[1] / (B * N);
    const int HW = out_size / (2 * B);
    int res = 1;
    while (res * res < HW) res++;   // integer sqrt (HW = res*res)

    // Workspace layout (floats): payload B*N*4 | feat2 B*N*2 | bbox B*4  (~0.8 MB)
    float*  ws      = (float*)d_ws;
    float4* payload = (float4*)ws;
    float2* feat2   = (float2*)(ws + (size_t)B * N * 4);
    float*  bbox    = ws + (size_t)B * N * 6;

    bbox_kernel<<<dim3(B), dim3(256), 0, stream>>>(coords, bbox, N);

    const int totalP = B * N;
    prep_kernel<<<dim3((totalP + 255) / 256), dim3(256), 0, stream>>>(
        coords, features, bbox, payload, feat2, N, C, res, totalP);

    const int tilesPerBatch = HW / 16;
    const int xBlocks = (tilesPerBatch + 7) / 8;   // 8 waves (tiles) per block
    ball_kernel<<<dim3(xBlocks, B), dim3(256), 0, stream>>>(
        payload, feat2, points_num, (float*)d_out, N, res, tilesPerBatch);
}